// MyModel_21363167330564
// MI455X (gfx1250) — compile-verified
//
#include <hip/hip_runtime.h>

typedef __attribute__((ext_vector_type(2))) float        v2f;
typedef __attribute__((ext_vector_type(8))) float        v8f;
typedef __attribute__((ext_vector_type(4))) unsigned int u32x4;
typedef __attribute__((ext_vector_type(8))) int          i32x8;
typedef __attribute__((ext_vector_type(4))) int          i32x4;

#define RG_DIM  64
#define RG_NREL 8
#define BPAD    68   // LDS row stride (floats): 64 + 4 pad -> conflict-free ds_load_b64

// -------------------------------------------------------------------------
// Kernel 1: WT[r][o][i] = sum_b coef[r,b] * bases[b][i][o]   (TRANSPOSED)
//           WT[8][o][i] = wself[i][o]
// Transposed storage makes every WMMA B-fragment ({W[k][n], W[k+1][n]}) a
// contiguous 8-byte pair in LDS -> single ds_load_b64, no repack movs.
// -------------------------------------------------------------------------
__global__ void rgcn_make_w(const float* __restrict__ bases,
                            const float* __restrict__ coef,
                            const float* __restrict__ wself,
                            float* __restrict__ WT) {
  int idx = blockIdx.x * blockDim.x + threadIdx.x;
  if (idx >= 9 * RG_DIM * RG_DIM) return;
  int r = idx >> 12;                 // /4096
  int o = (idx >> 6) & 63;
  int i = idx & 63;
  if (r < RG_NREL) {
    float acc = 0.f;
#pragma unroll
    for (int b = 0; b < 4; ++b)
      acc = fmaf(coef[r * 4 + b], bases[b * RG_DIM * RG_DIM + i * RG_DIM + o], acc);
    WT[idx] = acc;
  } else {
    WT[idx] = wself[i * RG_DIM + o];
  }
}

// -------------------------------------------------------------------------
// Kernel 2: per-relation pre-aggregation (aggregate BEFORE projecting):
//   S[etype[e], dst[e], :] += h[src[e], :]
// One wave32 per edge; each lane handles 2 of the 64 dims (float2).
// h (25.6 MB) stays L2-resident on MI455X (192 MB L2); atomics land in L2.
// -------------------------------------------------------------------------
__global__ void __launch_bounds__(256)
rgcn_scatter(const int* __restrict__ src, const int* __restrict__ dst,
             const int* __restrict__ etype, const float* __restrict__ h,
             float* __restrict__ S, int n_edges, int n_nodes) {
  long long t = (long long)blockIdx.x * blockDim.x + threadIdx.x;
  int e    = (int)(t >> 5);
  int lane = (int)(t & 31);
  if (e >= n_edges) return;
  int s = src[e];
  int d = dst[e];
  int r = etype[e];
  const float2 hv = *(const float2*)(h + (size_t)s * RG_DIM + lane * 2);
  float* base = S + ((size_t)r * n_nodes + d) * RG_DIM + lane * 2;
  atomicAdd(base + 0, hv.x);   // global_atomic_add_f32
  atomicAdd(base + 1, hv.y);
}

// -------------------------------------------------------------------------
// TDM: DMA one 64x64 fp32 weight tile (WT[r], row-major [o][i]) into LDS,
// with the TDM pad feature inserting 4 DWORDs after every 64 DWORDs so the
// LDS image has a 68-float row stride (bank-conflict-free B reads).
// Descriptor per CDNA5 ISA 8.3/8.4:
//   g0: count=1 | lds_addr | global_addr[56:0] | type=2
//   g1: data_size=4B, pad_enable, pad_interval=64dw, pad_amount=4dw,
//       tensor_dim0=64, tensor_dim1=64, tile_dim0=64, tile_dim1=64,
//       tensor_dim0_stride=64
// -------------------------------------------------------------------------
__device__ __forceinline__ void tdm_load_w(const float* WT, int r, unsigned lds_off) {
  unsigned long long ga =
      (unsigned long long)(uintptr_t)(WT + (size_t)r * RG_DIM * RG_DIM);
  u32x4 g0;
  g0[0] = 1u;                                            // count=1 (valid)
  g0[1] = lds_off;                                       // LDS byte address
  g0[2] = (unsigned)(ga & 0xffffffffu);                  // global_addr[31:0]
  g0[3] = (unsigned)((ga >> 32) & 0x01ffffffu)           // global_addr[56:32]
        | (2u << 30);                                    // type=2 ("image")
  i32x8 g1;
  g1[0] = (int)((2u << 16)      // data_size = 4 bytes
              | (1u << 20)      // pad_enable
              | (5u << 22)      // pad_interval: 64 DWORDs
              | (3u << 25));    // pad_amount:   4 DWORDs
  g1[1] = (int)(64u << 16);     // tensor_dim0 = 64 (bits 79:48)
  g1[2] = (int)(64u << 16);     // tensor_dim1 = 64 (bits 111:80)
  g1[3] = (int)(64u << 16);     // tile_dim0   = 64 (bits 127:112)
  g1[4] = 64;                   // tile_dim1   = 64, tile_dim2 = 0
  g1[5] = 64;                   // tensor_dim0_stride = 64
  g1[6] = 0;
  g1[7] = 0;
  i32x4 z = {0, 0, 0, 0};
#if __has_include(<hip/amd_detail/amd_gfx1250_TDM.h>)
  i32x8 z8 = {0, 0, 0, 0, 0, 0, 0, 0};
  __builtin_amdgcn_tensor_load_to_lds(g0, g1, z, z, z8, 0);   // clang-23 form
#else
  __builtin_amdgcn_tensor_load_to_lds(g0, g1, z, z, 0);       // ROCm 7.2 form
#endif
}

// -------------------------------------------------------------------------
// Kernel 3: batched WMMA GEMM:
//   out[n,:] = bias + sum_{r<8} S[r][n,:] @ W[r] + h[n,:] @ W[8]
// Each wave owns a 16-row x 64-col output tile: 4 v8f accumulators,
// K swept 4 at a time with v_wmma_f32_16x16x4_f32.  Weight tiles are
// double-buffered in LDS via the Tensor Data Mover: wave 0 DMAs W[r+1]
// while all waves run WMMAs on W[r]; s_wait_tensorcnt + barrier hand off.
// VGPR layouts per CDNA5 ISA 7.12.2:
//   A 16x4 f32 : lane l -> M=l%16, VGPRs {K0,K1}, K0=(l/16)*2
//   B 4x16 f32 : lane l -> N=l%16, VGPRs {K0,K1} rows, K0=(l/16)*2
//   C/D 16x16  : lane l -> N=l%16, VGPR v -> M=v+8*(l/16)
// -------------------------------------------------------------------------
__global__ void __launch_bounds__(256)
rgcn_gemm(const float* __restrict__ S, const float* __restrict__ h,
          const float* __restrict__ WT, const float* __restrict__ bias,
          float* __restrict__ out, int n_nodes) {
  __shared__ float wlds[2][RG_DIM * BPAD];   // 2 x 17 KB, TDM-padded layout

  const int wave = threadIdx.x >> 5;
  const int lane = threadIdx.x & 31;
  const int lrow = lane & 15;
  const int lhi  = lane >> 4;               // lane half: 0 or 1
  const int rowTile = blockIdx.x * 8 + wave;
  const int row0    = rowTile * 16;
  const bool active = (row0 < n_nodes);     // wave-uniform

  // Clamp A-row for ragged tail: output rows depend only on their own A row,
  // so clamped duplicate rows only affect outputs we never store.
  int arow = row0 + lrow;
  if (arow >= n_nodes) arow = n_nodes - 1;

  // Prime the pipeline: DMA W[0] into buffer 0 (wave 0 only; TDM ignores
  // EXEC and issues once per wave, so guard must be wave-uniform).
  if (wave == 0) {
    tdm_load_w(WT, 0, (unsigned)(uintptr_t)&wlds[0][0]);
    __builtin_amdgcn_s_wait_tensorcnt(0);
  }
  __syncthreads();

  // Accumulators initialized with bias (bias[n], n = c*16 + lrow).
  v8f acc[4];
#pragma unroll
  for (int c = 0; c < 4; ++c) {
    float bv = bias[c * 16 + lrow];
#pragma unroll
    for (int v = 0; v < 8; ++v) acc[c][v] = bv;
  }

  for (int r = 0; r < 9; ++r) {
    // Kick off the async DMA of the next weight tile into the other buffer.
    if (wave == 0 && r + 1 < 9)
      tdm_load_w(WT, r + 1, (unsigned)(uintptr_t)&wlds[(r + 1) & 1][0]);

    const float* buf = &wlds[r & 1][0];

    if (active) {      // wave-uniform: EXEC all-1s inside (WMMA requirement)
      const float* A =
          (r < RG_NREL) ? (S + (size_t)r * n_nodes * RG_DIM) : h;
      const float* Arow = A + (size_t)arow * RG_DIM + lhi * 2;

      // Prefetch next relation's A row (global_prefetch_b8).
      if (r + 1 < 9) {
        const float* An =
            (r + 1 < RG_NREL) ? (S + (size_t)(r + 1) * n_nodes * RG_DIM) : h;
        __builtin_prefetch(An + (size_t)arow * RG_DIM, 0, 0);
      }

#pragma unroll
      for (int kt = 0; kt < 16; ++kt) {          // K = 64, 4 per WMMA
        const int k0 = kt * 4;
        v2f a = *(const v2f*)(Arow + k0);        // global_load_b64
#pragma unroll
        for (int c = 0; c < 4; ++c) {            // 4 column tiles of 16
          // B fragment: contiguous pair WT[n][k0+lhi*2 .. +1] -> ds_load_b64
          v2f b = *(const v2f*)(buf + (c * 16 + lrow) * BPAD + k0 + lhi * 2);
          acc[c] = __builtin_amdgcn_wmma_f32_16x16x4_f32(
              false, a, false, b, (short)0, acc[c], false, false);
        }
      }
    }

    // Hand off: wave 0 waits for the DMA, then everyone crosses the barrier.
    if (wave == 0 && r + 1 < 9)
      __builtin_amdgcn_s_wait_tensorcnt(0);
    __syncthreads();
  }

  if (active) {
#pragma unroll
    for (int c = 0; c < 4; ++c) {
#pragma unroll
      for (int v = 0; v < 8; ++v) {
        int m = row0 + v + 8 * lhi;
        if (m < n_nodes)
          out[(size_t)m * RG_DIM + c * 16 + lrow] = acc[c][v];
      }
    }
  }
}

// -------------------------------------------------------------------------
// Host-side orchestration (graph-capture safe: only kernel launches and
// hipMemsetAsync on `stream`; all scratch in d_ws; no static state).
// -------------------------------------------------------------------------
extern "C" void kernel_launch(void* const* d_in, const int* in_sizes, int n_in,
                              void* d_out, int out_size, void* d_ws, size_t ws_size,
                              hipStream_t stream) {
  const int*   src    = (const int*)d_in[0];
  const int*   dst    = (const int*)d_in[1];
  const int*   ety    = (const int*)d_in[2];
  const float* emb    = (const float*)d_in[3];
  const float* bases1 = (const float*)d_in[4];
  const float* coef1  = (const float*)d_in[5];
  const float* wself1 = (const float*)d_in[6];
  const float* bias1  = (const float*)d_in[7];
  const float* bases2 = (const float*)d_in[8];
  const float* coef2  = (const float*)d_in[9];
  const float* wself2 = (const float*)d_in[10];
  const float* bias2  = (const float*)d_in[11];

  const int E = in_sizes[0];
  const int N = in_sizes[3] / RG_DIM;

  // Workspace layout (fp32):
  //   S  : [8, N, 64]   per-relation aggregated sums   (204.8 MB @ N=100k)
  //   h1 : [N, 64]      layer-1 output                 (25.6 MB)
  //   Wb : [9, 64, 64]  transposed combined weights    (144 KB)
  float* S  = (float*)d_ws;
  float* h1 = S  + (size_t)RG_NREL * N * RG_DIM;
  float* Wb = h1 + (size_t)N * RG_DIM;

  const size_t s_bytes = (size_t)RG_NREL * N * RG_DIM * sizeof(float);
  const int wblocks  = (9 * RG_DIM * RG_DIM + 255) / 256;
  const int sblocks  = (int)(((size_t)E * 32 + 255) / 256);
  const int rowTiles = (N + 15) / 16;
  const int gblocks  = (rowTiles + 7) / 8;

  // ---- Layer 1 ----
  rgcn_make_w<<<wblocks, 256, 0, stream>>>(bases1, coef1, wself1, Wb);
  hipMemsetAsync(S, 0, s_bytes, stream);
  rgcn_scatter<<<sblocks, 256, 0, stream>>>(src, dst, ety, emb, S, E, N);
  rgcn_gemm<<<gblocks, 256, 0, stream>>>(S, emb, Wb, bias1, h1, N);

  // ---- Layer 2 ----
  rgcn_make_w<<<wblocks, 256, 0, stream>>>(bases2, coef2, wself2, Wb);
  hipMemsetAsync(S, 0, s_bytes, stream);
  rgcn_scatter<<<sblocks, 256, 0, stream>>>(src, dst, ety, h1, S, E, N);
  rgcn_gemm<<<gblocks, 256, 0, stream>>>(S, h1, Wb, bias2, (float*)d_out, N);
}